// TensorProduct_75883482186070
// MI455X (gfx1250) — compile-verified
//
#include <hip/hip_runtime.h>
#include <cstdint>
#include <cstddef>

// One wave32 per row; each lane owns 4 consecutive channels (u = lane*4 .. lane*4+3).
// All global traffic is 128-bit, wave-coalesced, with non-temporal hints (pure stream).
// x2 (16 B/row) is staged block-wide into LDS via CDNA5 async global->LDS DMA.

typedef float v4f __attribute__((ext_vector_type(4)));

#define ROWS_PER_BLOCK 8   // 8 waves * 32 lanes = 256 threads

__global__ __launch_bounds__(256) void tp_l1x0e_kernel(
    const float* __restrict__ x1,     // (N, 512)
    const float* __restrict__ x2,     // (N, 4)
    const float* __restrict__ wts,    // (N, 640)
    float*       __restrict__ out,    // (N, 512)
    int nrows)
{
    const float S    = 0.088388347648318447f;  // 1/sqrt(128)
    const float ISQ3 = 0.57735026918962576f;   // 1/sqrt(3)
    const float ISQ2 = 0.70710678118654752f;   // 1/sqrt(2)

    __shared__ __align__(16) float sx2[ROWS_PER_BLOCK * 4];

    const int tid  = (int)threadIdx.x;
    const int wave = tid >> 5;
    const int lane = tid & 31;
    const int row0 = (int)blockIdx.x * ROWS_PER_BLOCK;

    // --- Stage this block's x2 rows (8 rows x 16B) into LDS with async DMA ---
    if (wave == 0) {
        if (lane < ROWS_PER_BLOCK && (row0 + lane) < nrows) {
            uint64_t gaddr = (uint64_t)(uintptr_t)(x2 + (size_t)(row0 + lane) * 4);
            uint32_t laddr = (uint32_t)(uintptr_t)(&sx2[lane * 4]);
            asm volatile("global_load_async_to_lds_b128 %0, %1, off"
                         :: "v"(laddr), "v"(gaddr) : "memory");
        }
        asm volatile("s_wait_asynccnt 0" ::: "memory");
    }
    __syncthreads();

    const int n = row0 + wave;
    if (n >= nrows) return;

    const v4f y   = *(const v4f*)(&sx2[wave * 4]);   // ds_load_b128
    const float y0  = y.x;
    const float y1a = y.y;
    const float y1b = y.z;
    const float y1c = y.w;

    const size_t xbase = (size_t)n * 512 + (size_t)lane * 4;  // x1 & out layout
    const size_t wbase = (size_t)n * 640 + (size_t)lane * 4;  // weights layout

    // Streamed, non-temporal b128 loads (no reuse -> keep L2 clean).
    const v4f x10  = __builtin_nontemporal_load((const v4f*)(x1 + xbase));
    const v4f x11a = __builtin_nontemporal_load((const v4f*)(x1 + xbase + 128));
    const v4f x11b = __builtin_nontemporal_load((const v4f*)(x1 + xbase + 256));
    const v4f x11c = __builtin_nontemporal_load((const v4f*)(x1 + xbase + 384));

    const v4f w0 = __builtin_nontemporal_load((const v4f*)(wts + wbase));
    const v4f w1 = __builtin_nontemporal_load((const v4f*)(wts + wbase + 128));
    const v4f w2 = __builtin_nontemporal_load((const v4f*)(wts + wbase + 256));
    const v4f w3 = __builtin_nontemporal_load((const v4f*)(wts + wbase + 384));
    const v4f w4 = __builtin_nontemporal_load((const v4f*)(wts + wbase + 512));

    // Scalar (l=0) x scalar (l=0) path
    const v4f dot = x11a * y1a + x11b * y1b + x11c * y1c;       // sum_i x1_1[i,u]*y1[i]
    const v4f o0  = (w0 * (x10 * y0) + (w3 * dot) * ISQ3) * S;

    // Vector path: cross[k] = (x1_1[:,u] x y1)[k]
    const v4f c0 = x11b * y1c - x11c * y1b;
    const v4f c1 = x11c * y1a - x11a * y1c;
    const v4f c2 = x11a * y1b - x11b * y1a;

    const v4f o1a = (w1 * (x10 * y1a) + w2 * (x11a * y0) + (w4 * c0) * ISQ2) * S;
    const v4f o1b = (w1 * (x10 * y1b) + w2 * (x11b * y0) + (w4 * c1) * ISQ2) * S;
    const v4f o1c = (w1 * (x10 * y1c) + w2 * (x11c * y0) + (w4 * c2) * ISQ2) * S;

    // Non-temporal b128 stores: row = [out0 | out1[0] | out1[1] | out1[2]]
    __builtin_nontemporal_store(o0,  (v4f*)(out + xbase));
    __builtin_nontemporal_store(o1a, (v4f*)(out + xbase + 128));
    __builtin_nontemporal_store(o1b, (v4f*)(out + xbase + 256));
    __builtin_nontemporal_store(o1c, (v4f*)(out + xbase + 384));
}

extern "C" void kernel_launch(void* const* d_in, const int* in_sizes, int n_in,
                              void* d_out, int out_size, void* d_ws, size_t ws_size,
                              hipStream_t stream) {
    const float* x1  = (const float*)d_in[0];   // (N,512) f32
    const float* x2  = (const float*)d_in[1];   // (N,4)   f32
    const float* wts = (const float*)d_in[2];   // (N,640) f32
    float* out = (float*)d_out;                 // (N,512) f32

    const int nrows  = in_sizes[1] / 4;
    const int blocks = (nrows + ROWS_PER_BLOCK - 1) / ROWS_PER_BLOCK;

    tp_l1x0e_kernel<<<blocks, 256, 0, stream>>>(x1, x2, wts, out, nrows);
}